// Y_prime_decoder_12137577578917
// MI455X (gfx1250) — compile-verified
//
#include <hip/hip_runtime.h>
#include <math.h>

typedef __attribute__((ext_vector_type(2))) float v2f;
typedef __attribute__((ext_vector_type(8))) float v8f;

#define D_HID 128

// ---------------- degree / normalization ----------------
__global__ void k_deg_init(float* __restrict__ deg, int n) {
  int i = blockIdx.x * blockDim.x + threadIdx.x;
  if (i < n) deg[i] = 1.0f;  // self-loop contributes 1
}

__global__ void k_deg_count(const int* __restrict__ dst, float* __restrict__ deg, int E) {
  int e = blockIdx.x * blockDim.x + threadIdx.x;
  if (e < E) atomicAdd(&deg[dst[e]], 1.0f);
}

__global__ void k_dinv(float* __restrict__ deg, int n) {
  int i = blockIdx.x * blockDim.x + threadIdx.x;
  if (i < n) deg[i] = rsqrtf(deg[i]);  // deg >= 1 always (self-loops)
}

// ---------------- GEMM1: H = X @ W1 via V_WMMA_F32_16X16X4_F32 ----------------
// Persistent waves: wave w holds ALL W1 fragments for its 16-col tile in
// registers (32 x v2f = 64 VGPRs), then grid-strides over 16-row tiles.
// Inner loop per tile: 32x global_load_b64 (A) + 32x WMMA + 8 stores.
__global__ void __launch_bounds__(256) k_gemm1(const float* __restrict__ X,
                                               const float* __restrict__ W,
                                               float* __restrict__ H, int nTiles) {
  const int wave = threadIdx.x >> 5;
  const int lane = threadIdx.x & 31;
  const int m    = lane & 15;
  const int half = lane >> 4;
  const int colBase = wave * 16;

  // B fragments for k = 0..127: bfrag[i] covers K-step k0 = 4*i.
  // Lane layout (ISA 7.12.2, 32-bit B 4x16): lane m holds N=colBase+m,
  // K = k0 + half*2 + {0,1}.
  const float* bcol = W + colBase + m;
  v2f bfrag[32];
#pragma unroll
  for (int i = 0; i < 32; ++i) {
    int kb = i * 4 + half * 2;
    bfrag[i] = v2f{bcol[(size_t)kb * D_HID], bcol[(size_t)(kb + 1) * D_HID]};
  }

  for (int tile = blockIdx.x; tile < nTiles; tile += gridDim.x) {
    const size_t rowBase = (size_t)tile * 16;
    // A fragment: lane m holds row rowBase+m, K = k0 + half*2 + {0,1} (8B loads)
    const float2* arow = (const float2*)(X + (rowBase + m) * D_HID + half * 2);

    v8f acc0 = {};
    v8f acc1 = {};
#pragma unroll
    for (int i = 0; i < 32; i += 2) {
      {
        float2 a = arow[i * 2];
        v2f av = {a.x, a.y};
        acc0 = __builtin_amdgcn_wmma_f32_16x16x4_f32(false, av, false, bfrag[i],
                                                     (short)0, acc0, false, false);
      }
      {
        float2 a = arow[i * 2 + 2];
        v2f av = {a.x, a.y};
        acc1 = __builtin_amdgcn_wmma_f32_16x16x4_f32(false, av, false, bfrag[i + 1],
                                                     (short)0, acc1, false, false);
      }
    }
    v8f acc = acc0 + acc1;

    // C layout: VGPR j -> row (j + half*8), col = colBase + m
    float* orow = H + (rowBase + half * 8) * D_HID + colBase + m;
#pragma unroll
    for (int j = 0; j < 8; ++j) orow[(size_t)j * D_HID] = acc[j];
  }
}

// ---------------- aggregation 1 ----------------
// init: out = b1 + dinv^2 * H1 (self-loop term), fully overwrites buffer
__global__ void k_agg1_init(const float* __restrict__ H1, const float* __restrict__ dinv,
                            const float* __restrict__ b1, float* __restrict__ out, int total) {
  int t = blockIdx.x * blockDim.x + threadIdx.x;
  if (t < total) {
    int node = t >> 7;
    int c    = t & 127;
    float di = dinv[node];
    out[t] = b1[c] + di * di * H1[t];
  }
}

// scatter: one wave per edge, lane handles 4 contiguous channels (float4 gather)
__global__ void __launch_bounds__(256) k_agg1_scatter(const float* __restrict__ H1,
                                                      const float* __restrict__ dinv,
                                                      const int* __restrict__ src,
                                                      const int* __restrict__ dst,
                                                      float* __restrict__ out, int E) {
  int wid  = (int)((blockIdx.x * blockDim.x + threadIdx.x) >> 5);
  int lane = threadIdx.x & 31;
  if (wid >= E) return;
  int s = src[wid];
  int d = dst[wid];
  float norm = dinv[s] * dinv[d];
  float4 v = ((const float4*)(H1 + (size_t)s * D_HID))[lane];
  float* o = out + (size_t)d * D_HID + lane * 4;
  atomicAdd(o + 0, v.x * norm);
  atomicAdd(o + 1, v.y * norm);
  atomicAdd(o + 2, v.z * norm);
  atomicAdd(o + 3, v.w * norm);
}

// ---------------- GEMM2 (128 -> 2) + self-loop init of logits ----------------
// one wave per node; lane handles 4 k's; shfl tree reduction
__global__ void __launch_bounds__(256) k_gemm2(const float* __restrict__ Hagg,
                                               const float* __restrict__ W2,
                                               const float* __restrict__ b2,
                                               const float* __restrict__ dinv,
                                               float* __restrict__ ylin,
                                               float* __restrict__ out2, int n) {
  int node = (int)((blockIdx.x * blockDim.x + threadIdx.x) >> 5);
  int lane = threadIdx.x & 31;
  if (node >= n) return;
  float4 h = ((const float4*)(Hagg + (size_t)node * D_HID))[lane];
  const float* wp = W2 + lane * 8;  // rows k=4*lane..4*lane+3, 2 cols each
  float p0 = h.x * wp[0] + h.y * wp[2] + h.z * wp[4] + h.w * wp[6];
  float p1 = h.x * wp[1] + h.y * wp[3] + h.z * wp[5] + h.w * wp[7];
#pragma unroll
  for (int off = 16; off > 0; off >>= 1) {
    p0 += __shfl_down(p0, off, 32);
    p1 += __shfl_down(p1, off, 32);
  }
  if (lane == 0) {
    ylin[node * 2 + 0] = p0;
    ylin[node * 2 + 1] = p1;
    float di = dinv[node];
    out2[node * 2 + 0] = b2[0] + di * di * p0;
    out2[node * 2 + 1] = b2[1] + di * di * p1;
  }
}

// ---------------- aggregation 2: one thread per edge ----------------
__global__ void k_agg2_scatter(const float* __restrict__ ylin, const float* __restrict__ dinv,
                               const int* __restrict__ src, const int* __restrict__ dst,
                               float* __restrict__ out, int E) {
  int e = blockIdx.x * blockDim.x + threadIdx.x;
  if (e < E) {
    int s = src[e];
    int d = dst[e];
    float norm = dinv[s] * dinv[d];
    atomicAdd(&out[d * 2 + 0], ylin[s * 2 + 0] * norm);
    atomicAdd(&out[d * 2 + 1], ylin[s * 2 + 1] * norm);
  }
}

// ---------------- stable 2-way softmax, in place ----------------
__global__ void k_softmax(float* __restrict__ out, int n) {
  int i = blockIdx.x * blockDim.x + threadIdx.x;
  if (i < n) {
    float a = out[2 * i], b = out[2 * i + 1];
    float mx = fmaxf(a, b);
    float ea = __expf(a - mx);
    float eb = __expf(b - mx);
    float inv = 1.0f / (ea + eb);
    out[2 * i]     = ea * inv;
    out[2 * i + 1] = eb * inv;
  }
}

extern "C" void kernel_launch(void* const* d_in, const int* in_sizes, int n_in,
                              void* d_out, int out_size, void* d_ws, size_t ws_size,
                              hipStream_t stream) {
  const float* X  = (const float*)d_in[0];
  const int*   ei = (const int*)d_in[1];
  const float* W1 = (const float*)d_in[2];
  const float* b1 = (const float*)d_in[3];
  const float* W2 = (const float*)d_in[4];
  const float* b2 = (const float*)d_in[5];
  float* out = (float*)d_out;

  const int N = in_sizes[0] / D_HID;   // 100000
  const int E = in_sizes[1] / 2;       // 1600000
  const int* src = ei;
  const int* dst = ei + E;

  // workspace layout (floats): dinv[N] | H1[N*128] | Hagg[N*128] | ylin[N*2]
  float* ws   = (float*)d_ws;
  float* dinv = ws;
  float* H1   = ws + (((size_t)N + 3) & ~(size_t)3);
  float* Hagg = H1 + (size_t)N * D_HID;
  float* ylin = Hagg + (size_t)N * D_HID;

  const int B = 256;
  // degrees + inverse sqrt
  k_deg_init<<<(N + B - 1) / B, B, 0, stream>>>(dinv, N);
  k_deg_count<<<(E + B - 1) / B, B, 0, stream>>>(dst, dinv, E);
  k_dinv<<<(N + B - 1) / B, B, 0, stream>>>(dinv, N);
  // layer 1: transform (persistent-wave WMMA f32), then aggregate
  int nTiles = N / 16;                       // 6250 (N is a multiple of 16)
  int gemmBlocks = nTiles < 1024 ? nTiles : 1024;
  k_gemm1<<<gemmBlocks, B, 0, stream>>>(X, W1, H1, nTiles);
  k_agg1_init<<<(N * D_HID + B - 1) / B, B, 0, stream>>>(H1, dinv, b1, Hagg, N * D_HID);
  k_agg1_scatter<<<(E + 7) / 8, B, 0, stream>>>(H1, dinv, src, dst, Hagg, E);
  // layer 2: transform (wave dot), then aggregate, then softmax
  k_gemm2<<<(N + 7) / 8, B, 0, stream>>>(Hagg, W2, b2, dinv, ylin, out, N);
  k_agg2_scatter<<<(E + B - 1) / B, B, 0, stream>>>(ylin, dinv, src, dst, out, E);
  k_softmax<<<(N + B - 1) / B, B, 0, stream>>>(out, N);
}